// HPNETLoss_16381005267478
// MI455X (gfx1250) — compile-verified
//
#include <hip/hip_runtime.h>
#include <hip/hip_bf16.h>

typedef __attribute__((ext_vector_type(2))) float v2f;
typedef __attribute__((ext_vector_type(8))) float v8f;

// ---------------------------------------------------------------------------
// Wave-level f32 sum using V_WMMA_F32_16X16X4_F32 with B = ones.
// A layout (16x4 f32): lanes 0-15 -> K=0,1 ; lanes 16-31 -> K=2,3 ; M = lane%16.
// We place each lane's value in its first A VGPR (K=0 / K=2), zero the other.
// D[m][n] = v[m] + v[m+16] (row sums, replicated over N). A lane's 8 C regs
// cover 8 distinct rows; summing them and adding the other half-wave's sum via
// shfl_xor(16) yields the full 32-lane total in every lane.
// EXEC must be all ones at the call site (callers guarantee reconvergence).
// ---------------------------------------------------------------------------
__device__ __forceinline__ float wave_sum_wmma(float v) {
    v2f a; a.x = v;    a.y = 0.0f;
    v2f b; b.x = 1.0f; b.y = 1.0f;
    v8f c = {};
    c = __builtin_amdgcn_wmma_f32_16x16x4_f32(false, a, false, b, (short)0, c,
                                              false, false);
    float s = c[0] + c[1] + c[2] + c[3] + c[4] + c[5] + c[6] + c[7];
    s += __shfl_xor(s, 16, 32);
    return s;
}

// ---------------------------------------------------------------------------
// Kernel 1: confidence loss partials. Pure HBM streaming (b128 loads) of the
// three 16.7M-element f32 arrays; per-thread FMA accumulation; WMMA wave
// reduction; one block partial per workgroup.
// ---------------------------------------------------------------------------
__global__ __launch_bounds__(256) void conf_partial_kernel(
    const float* __restrict__ conf, const float* __restrict__ conf_gt,
    const float* __restrict__ weight, float* __restrict__ partial,
    int nvec, int total) {
    const float4* __restrict__ c4 = (const float4*)conf;
    const float4* __restrict__ g4 = (const float4*)conf_gt;
    const float4* __restrict__ w4 = (const float4*)weight;

    const int tid    = blockIdx.x * blockDim.x + threadIdx.x;
    const int stride = gridDim.x * blockDim.x;

    float acc = 0.0f;
    for (int i = tid; i < nvec; i += stride) {
        float4 c = c4[i];
        float4 g = g4[i];
        float4 w = w4[i];
        float d0 = c.x - g.x, d1 = c.y - g.y, d2 = c.z - g.z, d3 = c.w - g.w;
        acc = fmaf(w.x * d0, d0, acc);
        acc = fmaf(w.y * d1, d1, acc);
        acc = fmaf(w.z * d2, d2, acc);
        acc = fmaf(w.w * d3, d3, acc);
    }
    // Scalar tail (empty for the reference shapes; kept for robustness).
    for (int i = nvec * 4 + tid; i < total; i += stride) {
        float d = conf[i] - conf_gt[i];
        acc = fmaf(weight[i] * d, d, acc);
    }

    // Reconverged here: EXEC all ones for the WMMA reduce.
    float wsum = wave_sum_wmma(acc);

    __shared__ float lds[8];
    const int lane = threadIdx.x & 31;
    const int wid  = threadIdx.x >> 5;
    if (lane == 0) lds[wid] = wsum;
    __syncthreads();
    if (threadIdx.x == 0) {
        float t = 0.0f;
        #pragma unroll
        for (int k = 0; k < 8; ++k) t += lds[k];
        partial[blockIdx.x] = t;
    }
}

// ---------------------------------------------------------------------------
// Kernel 2: depth + rotation loss partials over N=8192 annotations.
// ---------------------------------------------------------------------------
__device__ __forceinline__ void quat2mat9(float q0, float q1, float q2, float q3,
                                          float m[9]) {
    m[0] = q0 * q0 + q1 * q1 - q2 * q2 - q3 * q3;
    m[1] = 2.0f * (q1 * q2 - q0 * q3);
    m[2] = 2.0f * (q1 * q3 + q0 * q2);
    m[3] = 2.0f * (q1 * q2 + q0 * q3);
    m[4] = q0 * q0 - q1 * q1 + q2 * q2 - q3 * q3;
    m[5] = 2.0f * (q2 * q3 - q0 * q1);
    m[6] = 2.0f * (q1 * q3 - q0 * q2);
    m[7] = 2.0f * (q2 * q3 + q0 * q1);
    m[8] = q0 * q0 - q1 * q1 - q2 * q2 + q3 * q3;
}

__global__ __launch_bounds__(256) void ann_partial_kernel(
    const float* __restrict__ dr, const float* __restrict__ av,
    const int* __restrict__ flags, float* __restrict__ dpart,
    float* __restrict__ rpart, int N) {
    const int i = blockIdx.x * blockDim.x + threadIdx.x;

    float dterm = 0.0f, rterm = 0.0f;
    if (i < N) {
        const float  mask = (flags[i] != 0) ? 1.0f : 0.0f;
        const float* d = dr + (size_t)i * 5;
        const float* a = av + (size_t)i * 5;

        float dd = d[0] - a[0];
        dterm = mask * dd * dd;

        // m_pred from ann quaternion (reference does NOT normalize it).
        float p[9];
        quat2mat9(a[1], a[2], a[3], a[4], p);

        // m_gt from normalized predicted quaternion.
        float q0 = d[1], q1 = d[2], q2 = d[3], q3 = d[4];
        float rn = 1.0f / sqrtf(q0 * q0 + q1 * q1 + q2 * q2 + q3 * q3);
        float g[9];
        quat2mat9(q0 * rn, q1 * rn, q2 * rn, q3 * rn, g);

        // RY = diag(-1, 1, -1): m_pred @ RY scales columns by (-1, 1, -1).
        float n1 = 0.0f, n2 = 0.0f;
        const float s[3] = {-1.0f, 1.0f, -1.0f};
        #pragma unroll
        for (int r = 0; r < 3; ++r) {
            #pragma unroll
            for (int cI = 0; cI < 3; ++cI) {
                float gv = g[r * 3 + cI];
                float pv = p[r * 3 + cI];
                float e1 = gv - pv;
                float e2 = gv - pv * s[cI];
                n1 = fmaf(e1, e1, n1);
                n2 = fmaf(e2, e2, n2);
            }
        }
        rterm = mask * fminf(sqrtf(n1), sqrtf(n2));
    }

    // Reconverged: EXEC all ones for the WMMA reduces.
    float wd = wave_sum_wmma(dterm);
    float wr = wave_sum_wmma(rterm);

    __shared__ float ldsd[8];
    __shared__ float ldsr[8];
    const int lane = threadIdx.x & 31;
    const int wid  = threadIdx.x >> 5;
    if (lane == 0) { ldsd[wid] = wd; ldsr[wid] = wr; }
    __syncthreads();
    if (threadIdx.x == 0) {
        float td = 0.0f, tr = 0.0f;
        #pragma unroll
        for (int k = 0; k < 8; ++k) { td += ldsd[k]; tr += ldsr[k]; }
        dpart[blockIdx.x] = td;
        rpart[blockIdx.x] = tr;
    }
}

// ---------------------------------------------------------------------------
// Kernel 3: one wave. Chain WMMAs (C accumulates across iterations) to sum the
// confidence partials 64-at-a-time, then WMMA-reduce the tiny ann partials.
// ---------------------------------------------------------------------------
__global__ __launch_bounds__(32) void final_reduce_kernel(
    const float* __restrict__ cpart, const float* __restrict__ dpart,
    const float* __restrict__ rpart, float* __restrict__ out,
    int nconf, int nann, float conf_scale, float ann_scale) {
    const int lane = threadIdx.x;  // single wave32

    v2f b; b.x = 1.0f; b.y = 1.0f;
    v8f c = {};
    const int iters = nconf >> 6;  // nconf is a multiple of 64
    for (int i = 0; i < iters; ++i) {
        const int base = i * 64;
        v2f a;
        a.x = cpart[base + lane];
        a.y = cpart[base + 32 + lane];
        c = __builtin_amdgcn_wmma_f32_16x16x4_f32(false, a, false, b, (short)0,
                                                  c, false, false);
    }
    float s = c[0] + c[1] + c[2] + c[3] + c[4] + c[5] + c[6] + c[7];
    s += __shfl_xor(s, 16, 32);
    const float conf_total = s;

    float dv = 0.0f, rv = 0.0f;
    if (lane < nann) { dv = dpart[lane]; rv = rpart[lane]; }
    float dt = wave_sum_wmma(dv);
    float rt = wave_sum_wmma(rv);

    if (lane == 0) {
        out[0] = conf_total * conf_scale;  // / (256*256), per reference
        out[1] = dt * ann_scale;           // / N
        out[2] = rt * ann_scale;           // / N
    }
}

// ---------------------------------------------------------------------------
// Host launcher.
// Inputs (setup_inputs order):
//   0: confidence    (B,1,H,W) f32   1: confidence_gt (B,1,H,W) f32
//   2: weight        (B,H,W)   f32   3: depth_and_rotation (N,5) f32
//   4: ann_values    (N,5)     f32   5: ann_flags (N) bool -> int per harness
// Output: [confidence_loss, depth_loss, rotation_loss] f32.
// ---------------------------------------------------------------------------
extern "C" void kernel_launch(void* const* d_in, const int* in_sizes, int n_in,
                              void* d_out, int out_size, void* d_ws,
                              size_t ws_size, hipStream_t stream) {
    const float* conf    = (const float*)d_in[0];
    const float* conf_gt = (const float*)d_in[1];
    const float* weight  = (const float*)d_in[2];
    const float* dr      = (const float*)d_in[3];
    const float* av      = (const float*)d_in[4];
    const int*   flags   = (const int*)d_in[5];
    float*       out     = (float*)d_out;
    float*       ws      = (float*)d_ws;

    const int total = in_sizes[0];      // 256*256*256 = 16,777,216
    const int nvec  = total >> 2;       // float4 count
    const int N     = in_sizes[5];      // 8192

    const int CBLOCKS = 2048;           // conf partials: multiple of 64
    const int ABLOCKS = (N + 255) / 256;  // 32 (<= 32 lanes in final reduce)

    float* cpart = ws;                  // [0, CBLOCKS)
    float* dpart = ws + CBLOCKS;        // [CBLOCKS, CBLOCKS+32)
    float* rpart = ws + CBLOCKS + 32;   // [CBLOCKS+32, CBLOCKS+64)

    conf_partial_kernel<<<CBLOCKS, 256, 0, stream>>>(conf, conf_gt, weight,
                                                     cpart, nvec, total);
    ann_partial_kernel<<<ABLOCKS, 256, 0, stream>>>(dr, av, flags, dpart,
                                                    rpart, N);
    final_reduce_kernel<<<1, 32, 0, stream>>>(cpart, dpart, rpart, out,
                                              CBLOCKS, ABLOCKS,
                                              1.0f / 65536.0f,
                                              1.0f / (float)N);
}